// PointNet_SA_39213051413226
// MI455X (gfx1250) — compile-verified
//
#include <hip/hip_runtime.h>
#include <hip/hip_bf16.h>
#include <stdint.h>

typedef float v2f __attribute__((ext_vector_type(2)));
typedef float v8f __attribute__((ext_vector_type(8)));

#define B_  16
#define N_  4096
#define C_  64
#define S_  1024
#define K_  32
#define R2_ 0.04f   // radius 0.2 squared

// =====================================================================
// Kernel 1: farthest point sampling. One 1024-thread block per batch.
// Each thread keeps 4 points + running min-dist in registers.
// Argmax via (dist_bits<<32 | ~idx) key: max key == max dist, ties -> min idx
// (matches jnp.argmax first-occurrence semantics; dists are >= 0 so the
// float bit pattern is monotonic).
// =====================================================================
__global__ __launch_bounds__(1024)
void fps_kernel(const float* __restrict__ xyz,
                int* __restrict__ fpsIdx,
                float* __restrict__ newXyz) {
  const int b    = blockIdx.x;
  const int tid  = threadIdx.x;
  const int lane = tid & 31;
  const int wid  = tid >> 5;
  const float* bx = xyz + (size_t)b * N_ * 3;

  float px[4], py[4], pz[4], dd[4];
#pragma unroll
  for (int u = 0; u < 4; ++u) {
    int i = tid + u * 1024;
    px[u] = bx[i * 3 + 0];
    py[u] = bx[i * 3 + 1];
    pz[u] = bx[i * 3 + 2];
    dd[u] = 1e10f;
  }

  __shared__ unsigned long long warpKey[32];
  __shared__ int sFar;

  int far = 0;
  for (int it = 0; it < S_; ++it) {
    if (tid == 0) {
      fpsIdx[b * S_ + it] = far;
      newXyz[(b * S_ + it) * 3 + 0] = bx[far * 3 + 0];
      newXyz[(b * S_ + it) * 3 + 1] = bx[far * 3 + 1];
      newXyz[(b * S_ + it) * 3 + 2] = bx[far * 3 + 2];
    }
    const float cx = bx[far * 3 + 0];
    const float cy = bx[far * 3 + 1];
    const float cz = bx[far * 3 + 2];

    float bd = -1.0f; int bi = 0;
#pragma unroll
    for (int u = 0; u < 4; ++u) {
      float dx = px[u] - cx, dy = py[u] - cy, dz = pz[u] - cz;
      float d  = dx * dx + dy * dy + dz * dz;
      dd[u] = fminf(dd[u], d);
      if (dd[u] > bd) { bd = dd[u]; bi = tid + u * 1024; }
    }
    unsigned int fb = __float_as_uint(bd > 0.0f ? bd : 0.0f);
    unsigned long long key =
        ((unsigned long long)fb << 32) |
        (unsigned long long)(0xFFFFFFFFu - (unsigned)bi);

    // wave32 reduction
    for (int off = 16; off; off >>= 1) {
      unsigned long long o = __shfl_xor(key, off, 32);
      key = (o > key) ? o : key;
    }
    if (lane == 0) warpKey[wid] = key;
    __syncthreads();
    if (wid == 0) {
      unsigned long long k2 = warpKey[lane];
      for (int off = 16; off; off >>= 1) {
        unsigned long long o = __shfl_xor(k2, off, 32);
        k2 = (o > k2) ? o : k2;
      }
      if (lane == 0)
        sFar = (int)(0xFFFFFFFFu - (unsigned)(k2 & 0xFFFFFFFFull));
    }
    __syncthreads();
    far = sFar;
    __syncthreads();   // warpKey/sFar reuse next iteration
  }
}

// =====================================================================
// Kernel 2: ball query. One wave per (b,s) group; ballot+popc prefix
// compaction of the first K=32 in-radius indices in ascending order,
// padded with the first found index (0 if none), matching the reference.
// =====================================================================
__global__ __launch_bounds__(256)
void ball_kernel(const float* __restrict__ xyz,
                 const float* __restrict__ newXyz,
                 int* __restrict__ ballIdx) {
  const int wave = (int)((blockIdx.x * blockDim.x + threadIdx.x) >> 5);
  const int lane = threadIdx.x & 31;
  const int b = wave >> 10;
  const float* bx = xyz + (size_t)b * N_ * 3;
  const float cx = newXyz[wave * 3 + 0];
  const float cy = newXyz[wave * 3 + 1];
  const float cz = newXyz[wave * 3 + 2];
  int* out = ballIdx + (size_t)wave * K_;

  int count = 0, first = 0;
  bool haveFirst = false;
  for (int c = 0; c < N_ / 32; ++c) {
    int i = c * 32 + lane;
    float dx = bx[i * 3 + 0] - cx;
    float dy = bx[i * 3 + 1] - cy;
    float dz = bx[i * 3 + 2] - cz;
    bool inR = (dx * dx + dy * dy + dz * dz) <= R2_;
    unsigned int m = (unsigned int)__ballot(inR);
    if (!haveFirst && m) { first = c * 32 + (__ffs(m) - 1); haveFirst = true; }
    int prefix = __popc(m & ((1u << lane) - 1u));
    int pos = count + prefix;
    if (inR && pos < K_) out[pos] = i;
    count += __popc(m);
    if (count >= K_) break;
  }
  if (count < K_ && lane >= count && lane < K_) out[lane] = first;
}

// =====================================================================
// Kernel 3: gather + 3-layer MLP (f32 WMMA 16x16x4) + max over K.
// One wave per group; 2 waves per block. G = 32x68 in LDS (stride 68
// avoids 64-bank conflicts). Layer outputs ping-pong between two LDS
// tiles; layer 3 max-reduces D fragments directly to global.
// =====================================================================

// B fragment: W is KxF row-major. v0: lanes 0-15 hold row k0, lanes 16-31
// row k0+2; v1: rows k0+1 / k0+3 (ISA A/B half-lane striping).
template <int KVALID>
__device__ __forceinline__ v2f load_bfrag(const float* __restrict__ W, int F,
                                          int k0, int n0, int lane) {
  int n  = n0 + (lane & 15);
  int kb = k0 + ((lane >> 4) << 1);
  v2f r;
  r.x = (kb     < KVALID) ? W[kb * F + n]       : 0.0f;
  r.y = (kb + 1 < KVALID) ? W[(kb + 1) * F + n] : 0.0f;
  return r;
}

// A fragment from LDS tile (stride 68): lane&15 = row, lane>>4 selects K pair.
__device__ __forceinline__ v2f load_afrag(const float* base, int k0, int lane,
                                          int mtile) {
  int r = (lane & 15) + mtile * 16;
  const float* p = base + r * 68 + k0 + ((lane >> 4) << 1);
  v2f a; a.x = p[0]; a.y = p[1];
  return a;
}

template <int KSTEPS, int KVALID, int F>
__device__ __forceinline__ void mlp_layer(const float (*in)[68],
                                          float (*out)[68],
                                          const float* __restrict__ W,
                                          const float* __restrict__ bias,
                                          int lane) {
  const float* base = &in[0][0];
  for (int nt = 0; nt < F / 16; ++nt) {
    v8f acc0 = {}; v8f acc1 = {};
    const int n0 = nt * 16;
    for (int ks = 0; ks < KSTEPS; ++ks) {
      const int k0 = ks * 4;
      v2f bf = load_bfrag<KVALID>(W, F, k0, n0, lane);
      v2f a0 = load_afrag(base, k0, lane, 0);
      v2f a1 = load_afrag(base, k0, lane, 1);
      acc0 = __builtin_amdgcn_wmma_f32_16x16x4_f32(false, a0, false, bf,
                                                   (short)0, acc0, false, false);
      acc1 = __builtin_amdgcn_wmma_f32_16x16x4_f32(false, a1, false, bf,
                                                   (short)0, acc1, false, false);
    }
    const int nn = n0 + (lane & 15);
    const float bv = bias[nn];
    const int hi = lane >> 4;
#pragma unroll
    for (int r = 0; r < 8; ++r) {
      // D layout: VGPR r -> row r (lanes 0-15) / row r+8 (lanes 16-31)
      out[r + 8 * hi][nn]      = fmaxf(acc0[r] + bv, 0.0f);
      out[16 + r + 8 * hi][nn] = fmaxf(acc1[r] + bv, 0.0f);
    }
  }
}

__global__ __launch_bounds__(64)
void mlp_kernel(const float* __restrict__ xyz,
                const float* __restrict__ points,
                const float* __restrict__ w0, const float* __restrict__ b0,
                const float* __restrict__ w1, const float* __restrict__ b1,
                const float* __restrict__ w2, const float* __restrict__ b2,
                const float* __restrict__ newXyz,
                const int* __restrict__ ballIdx,
                float* __restrict__ outPts) {
  __shared__ float sA[2][32][68];
  __shared__ float sB[2][32][68];
  const int wv   = threadIdx.x >> 5;
  const int lane = threadIdx.x & 31;
  const int g    = blockIdx.x * 2 + wv;
  const int b    = g >> 10;

  // ---- gather: lane = sample row; 64 point channels + 3 relative xyz + pad
  const int pi = ballIdx[(size_t)g * K_ + lane];
  const float* src = points + ((size_t)b * N_ + pi) * C_;
  float* row = sA[wv][lane];
#pragma unroll
  for (int c = 0; c < C_; c += 4) {
    float4 v = *(const float4*)(src + c);
    row[c] = v.x; row[c + 1] = v.y; row[c + 2] = v.z; row[c + 3] = v.w;
  }
  const float cx = newXyz[g * 3 + 0];
  const float cy = newXyz[g * 3 + 1];
  const float cz = newXyz[g * 3 + 2];
  const float* pxyz = xyz + ((size_t)b * N_ + pi) * 3;
  row[64] = pxyz[0] - cx;
  row[65] = pxyz[1] - cy;
  row[66] = pxyz[2] - cz;
  row[67] = 0.0f;
  __syncthreads();

  // ---- layer 1: 67 -> 64 (K padded to 68, W0 rows >= 67 read as 0)
  mlp_layer<17, 67, 64>(sA[wv], sB[wv], w0, b0, lane);
  __syncthreads();
  // ---- layer 2: 64 -> 64
  mlp_layer<16, 64, 64>(sB[wv], sA[wv], w1, b1, lane);
  __syncthreads();

  // ---- layer 3: 64 -> 128, fused max over the 32 samples
  const float* base = &sA[wv][0][0];
  float* outp = outPts + (size_t)g * 128;
  for (int nt = 0; nt < 8; ++nt) {
    v8f acc0 = {}; v8f acc1 = {};
    const int n0 = nt * 16;
    for (int ks = 0; ks < 16; ++ks) {
      const int k0 = ks * 4;
      v2f bf = load_bfrag<64>(w2, 128, k0, n0, lane);
      v2f a0 = load_afrag(base, k0, lane, 0);
      v2f a1 = load_afrag(base, k0, lane, 1);
      acc0 = __builtin_amdgcn_wmma_f32_16x16x4_f32(false, a0, false, bf,
                                                   (short)0, acc0, false, false);
      acc1 = __builtin_amdgcn_wmma_f32_16x16x4_f32(false, a1, false, bf,
                                                   (short)0, acc1, false, false);
    }
    // column max over 32 rows: 8 regs x 2 M-tiles per lane, then lane-half fold
    float m = acc0[0];
#pragma unroll
    for (int r = 0; r < 8; ++r) {
      m = fmaxf(m, acc0[r]);
      m = fmaxf(m, acc1[r]);
    }
    m = fmaxf(m, __shfl_xor(m, 16, 32));
    const int nn = n0 + (lane & 15);
    const float v = fmaxf(m + b2[nn], 0.0f);   // relu(max+b) == max(relu(x+b))
    if (lane < 16) outp[nn] = v;
  }
}

// =====================================================================
extern "C" void kernel_launch(void* const* d_in, const int* in_sizes, int n_in,
                              void* d_out, int out_size, void* d_ws, size_t ws_size,
                              hipStream_t stream) {
  const float* xyz    = (const float*)d_in[0];
  const float* points = (const float*)d_in[1];
  const float* w0 = (const float*)d_in[2];
  const float* b0 = (const float*)d_in[3];
  const float* w1 = (const float*)d_in[4];
  const float* b1 = (const float*)d_in[5];
  const float* w2 = (const float*)d_in[6];
  const float* b2 = (const float*)d_in[7];

  float* newXyz = (float*)d_out;                       // B*S*3 floats
  float* newPts = (float*)d_out + B_ * S_ * 3;         // B*S*128 floats

  int* fpsIdx  = (int*)d_ws;                                           // B*S
  int* ballIdx = (int*)((char*)d_ws + ((B_ * S_ * sizeof(int) + 255u) & ~255u)); // B*S*K

  fps_kernel<<<B_, 1024, 0, stream>>>(xyz, fpsIdx, newXyz);
  ball_kernel<<<(B_ * S_ * 32) / 256, 256, 0, stream>>>(xyz, newXyz, ballIdx);
  mlp_kernel<<<(B_ * S_) / 2, 64, 0, stream>>>(xyz, points, w0, b0, w1, b1,
                                               w2, b2, newXyz, ballIdx, newPts);
}